// Embedding_88270167867496
// MI455X (gfx1250) — compile-verified
//
#include <hip/hip_runtime.h>
#include <hip/hip_bf16.h>

// out[b,s,e] = W_e[e, tokens[b,s]] + W_p[s,e]
//   tokens: (2, 2048) int32
//   W_e:    (1024, 32000) f32   (row stride V -> gather of column tokens[b,s])
//   W_p:    (2048, 1024)  f32
//   out:    (2, 2048, 1024) f32
//
// Memory-bound (~150 MB total traffic -> ~6.5us floor at 23.3 TB/s).
// CDNA5 path: W_p row staged to LDS via GLOBAL_LOAD_ASYNC_TO_LDS_B128
// (ASYNCcnt), reused for both batch entries sharing the same s.

#define EMB_E 1024
#define EMB_V 32000
#define EMB_S 2048

#if defined(__AMDGCN__) &&                                        \
    __has_builtin(__builtin_amdgcn_global_load_async_to_lds_b128) && \
    __has_builtin(__builtin_amdgcn_s_wait_asynccnt)
#define EMB_HAS_ASYNC_LDS 1
#else
#define EMB_HAS_ASYNC_LDS 0
#endif

#if EMB_HAS_ASYNC_LDS
typedef __attribute__((__vector_size__(4 * sizeof(int)))) int emb_v4i;
typedef __attribute__((address_space(1))) emb_v4i* emb_gptr_v4i;
typedef __attribute__((address_space(3))) emb_v4i* emb_lptr_v4i;
#endif

__global__ __launch_bounds__(256) void embed_gather_add_kernel(
    const int* __restrict__ tokens,   // (2, 2048)
    const float* __restrict__ We,     // (1024, 32000)
    const float* __restrict__ Wp,     // (2048, 1024)
    float* __restrict__ out)          // (2, 2048, 1024)
{
    __shared__ float wp_row[EMB_E];

    const int s   = blockIdx.x;       // 0..2047
    const int tid = threadIdx.x;      // 0..255

    // Uniform per-block token reads -> scalar loads.
    const int tok0 = tokens[s];           // b = 0
    const int tok1 = tokens[EMB_S + s];   // b = 1

    const float* wp = Wp + (size_t)s * EMB_E;

#if EMB_HAS_ASYNC_LDS
    // 256 threads x 16B = 4KB = full W_p row, async global->LDS (ASYNCcnt).
    __builtin_amdgcn_global_load_async_to_lds_b128(
        (emb_gptr_v4i)(wp + tid * 4),
        (emb_lptr_v4i)(wp_row + tid * 4),
        /*offset=*/0, /*cpol=*/0);
    __builtin_amdgcn_s_wait_asynccnt(0);
    __syncthreads();
#else
    ((float4*)wp_row)[tid] = ((const float4*)wp)[tid];
    __syncthreads();
#endif

    float* out0 = out + (size_t)s * EMB_E;                        // b = 0
    float* out1 = out + (size_t)(EMB_S + s) * EMB_E;              // b = 1

#pragma unroll
    for (int i = 0; i < 4; ++i) {
        const int e = tid + i * 256;
        // Gather: one 4B element per 128KB-strided row; L2-resident after
        // first touch (W_e = 128MB < 192MB L2).
        const float g0 = We[(size_t)e * EMB_V + tok0];
        const float g1 = We[(size_t)e * EMB_V + tok1];
        const float p  = wp_row[e];
        // Coalesced stores (lane-contiguous e within each wave).
        out0[e] = g0 + p;
        out1[e] = g1 + p;
    }
}

extern "C" void kernel_launch(void* const* d_in, const int* in_sizes, int n_in,
                              void* d_out, int out_size, void* d_ws, size_t ws_size,
                              hipStream_t stream) {
    (void)in_sizes; (void)n_in; (void)d_ws; (void)ws_size; (void)out_size;

    const int*   tokens = (const int*)d_in[0];    // (2, 2048)
    const float* We     = (const float*)d_in[1];  // (1024, 32000)
    const float* Wp     = (const float*)d_in[2];  // (2048, 1024)
    float*       out    = (float*)d_out;          // (2, 2048, 1024)

    embed_gather_add_kernel<<<EMB_S, 256, 0, stream>>>(tokens, We, Wp, out);
}